// LSTM_67980742362036
// MI455X (gfx1250) — compile-verified
//
#include <hip/hip_runtime.h>

typedef __attribute__((ext_vector_type(2))) float v2f;
typedef __attribute__((ext_vector_type(8))) float v8f;

#define TB 4096
#define TT 200
#define TI 32
#define TH 6
#define TG 24   // 4*H

__device__ __forceinline__ float sigm(float x) {
    return 1.0f / (1.0f + __expf(-x));
}
__device__ __forceinline__ float tanh_f(float x) {
    float e = __expf(-2.0f * x);
    return (1.0f - e) / (1.0f + e);
}

// D = A(16x4 f32) * B(4x16 f32) + C(16x16 f32)
__device__ __forceinline__ v8f wmma4(v2f a, v2f b, v8f c) {
    return __builtin_amdgcn_wmma_f32_16x16x4_f32(
        /*neg_a=*/false, a, /*neg_b=*/false, b,
        /*c_mod=*/(short)0, c, /*reuse_a=*/false, /*reuse_b=*/false);
}

// One wave (32 lanes) handles 16 batch elements across all T timesteps.
// Gates on the WMMA M axis (permuted so each half-wave owns a complete
// {i,f,g,o} set for 3 hidden units), batch (16) on N.
//
// Gate-row permutation G(tile, m):  hrow=m>>3, q=(m>>2)&1, type=m&3
//   tile0: gate = type*6 + 3*hrow + q          (k = 3*hrow + q)
//   tile1: gate = type*6 + 3*hrow + 2, valid iff q==0   (k = 3*hrow + 2)
// => lane (n, hi) reads from its accumulators the i/f/g/o of k = 3*hi + {0,1,2}.
//
// K permutations (GEMM sum order is free):
//   x GEMM, chunk kc: slots {K0,K1,K2,K3} = channels {2kc, 2kc+1, 16+2kc, 16+2kc+1}
//     => lane's B data = x[n][t][16*hi .. 16*hi+15], contiguous (4x b128).
//   h GEMM, chunk0: cols {0,1,3,4}; chunk1: cols {2,-,5,-}
//     => lane supplies only its local h[3*hi + kk], no cross-lane traffic.
__global__ __launch_bounds__(32)
void lstm_fused_kernel(const float* __restrict__ x,
                       const float* __restrict__ Wih0, const float* __restrict__ Whh0,
                       const float* __restrict__ bih0, const float* __restrict__ bhh0,
                       const float* __restrict__ Wih1, const float* __restrict__ Whh1,
                       const float* __restrict__ bih1, const float* __restrict__ bhh1,
                       const float* __restrict__ Wlin, const float* __restrict__ blin,
                       float* __restrict__ out)
{
    const int lane = threadIdx.x & 31;
    const int nloc = lane & 15;   // batch slot within wave (N) / tile row (A)
    const int hi   = lane >> 4;   // half-wave selector
    const int bglob = blockIdx.x * 16 + nloc;

    // Gate-row permutation for this lane's A-tile row (row index = nloc)
    const int hrow  = nloc >> 3;
    const int q     = (nloc >> 2) & 1;
    const int type  = nloc & 3;
    const int gate0 = type * 6 + 3 * hrow + q;        // tile0 row -> gate (always valid)
    const int gate1 = type * 6 + 3 * hrow + 2;        // tile1 row -> gate
    const float g1m = (q == 0) ? 1.0f : 0.0f;         // tile1 validity mask

    // ---------------- loop-invariant A-tiles (weights) ----------------
    // A 16x4 layout: lanes 0-15: M=lane, vgpr{0,1}=K{0,1}; lanes 16-31: M=lane-16, K{2,3}.
    v2f Aih0[2][8];          // W_ih0 [24x32]: 2 M-tiles x 8 K-chunks (permuted K)
    v2f Ahh0[2][2];          // W_hh0 [24x6]
    v2f Aih1[2][2];          // W_ih1 [24x6]
    v2f Ahh1[2][2];          // W_hh1 [24x6]

    #pragma unroll
    for (int kc = 0; kc < 8; ++kc) {
        const int col = kc * 2 + 16 * hi;             // channel for slot K=2*hi+j
        Aih0[0][kc].x = Wih0[gate0 * TI + col];
        Aih0[0][kc].y = Wih0[gate0 * TI + col + 1];
        Aih0[1][kc].x = Wih0[gate1 * TI + col]     * g1m;
        Aih0[1][kc].y = Wih0[gate1 * TI + col + 1] * g1m;
    }
    {
        const int c0 = 3 * hi;                        // chunk0 cols: 3*hi, 3*hi+1
        const int c2 = 3 * hi + 2;                    // chunk1 col:  3*hi+2
        Ahh0[0][0].x = Whh0[gate0 * TH + c0];
        Ahh0[0][0].y = Whh0[gate0 * TH + c0 + 1];
        Ahh0[0][1].x = Whh0[gate0 * TH + c2];
        Ahh0[0][1].y = 0.0f;
        Ahh0[1][0].x = Whh0[gate1 * TH + c0]     * g1m;
        Ahh0[1][0].y = Whh0[gate1 * TH + c0 + 1] * g1m;
        Ahh0[1][1].x = Whh0[gate1 * TH + c2]     * g1m;
        Ahh0[1][1].y = 0.0f;

        Aih1[0][0].x = Wih1[gate0 * TH + c0];
        Aih1[0][0].y = Wih1[gate0 * TH + c0 + 1];
        Aih1[0][1].x = Wih1[gate0 * TH + c2];
        Aih1[0][1].y = 0.0f;
        Aih1[1][0].x = Wih1[gate1 * TH + c0]     * g1m;
        Aih1[1][0].y = Wih1[gate1 * TH + c0 + 1] * g1m;
        Aih1[1][1].x = Wih1[gate1 * TH + c2]     * g1m;
        Aih1[1][1].y = 0.0f;

        Ahh1[0][0].x = Whh1[gate0 * TH + c0];
        Ahh1[0][0].y = Whh1[gate0 * TH + c0 + 1];
        Ahh1[0][1].x = Whh1[gate0 * TH + c2];
        Ahh1[0][1].y = 0.0f;
        Ahh1[1][0].x = Whh1[gate1 * TH + c0]     * g1m;
        Ahh1[1][0].y = Whh1[gate1 * TH + c0 + 1] * g1m;
        Ahh1[1][1].x = Whh1[gate1 * TH + c2]     * g1m;
        Ahh1[1][1].y = 0.0f;
    }

    // Biases in C layout: acc0 row r <-> gate (r&3)*6 + 3*hi + (r>>2);
    //                     acc1 row r (r<4) <-> gate r*6 + 3*hi + 2.
    float b0a[8], b0b[8], b1a[8], b1b[8];
    #pragma unroll
    for (int r = 0; r < 8; ++r) {
        const int ga = (r & 3) * 6 + 3 * hi + (r >> 2);
        b0a[r] = bih0[ga] + bhh0[ga];
        b1a[r] = bih1[ga] + bhh1[ga];
        if (r < 4) {
            const int gb = r * 6 + 3 * hi + 2;
            b0b[r] = bih0[gb] + bhh0[gb];
            b1b[r] = bih1[gb] + bhh1[gb];
        } else {
            b0b[r] = 0.0f;
            b1b[r] = 0.0f;
        }
    }

    // Lane-local LSTM state: this lane owns hidden units k = 3*hi + {0,1,2}
    float h1l[3], c1l[3], h2l[3], c2l[3];
    #pragma unroll
    for (int k = 0; k < 3; ++k) { h1l[k] = 0.f; c1l[k] = 0.f; h2l[k] = 0.f; c2l[k] = 0.f; }

    // x stream: lane reads 16 contiguous floats per step, double-buffered.
    const float* xptr = x + (size_t)bglob * TT * TI + 16 * hi;
    float xc[16], xn[16];
    #pragma unroll
    for (int qd = 0; qd < 4; ++qd) {
        const float4 v = *(const float4*)(xptr + 4 * qd);
        xc[4*qd+0] = v.x; xc[4*qd+1] = v.y; xc[4*qd+2] = v.z; xc[4*qd+3] = v.w;
    }

    #pragma unroll 1
    for (int t = 0; t < TT; ++t) {
        // prefetch next timestep (clamped; result used only at loop end)
        const float* xnp = xptr + (size_t)((t + 1 < TT) ? (t + 1) : t) * TI;
        #pragma unroll
        for (int qd = 0; qd < 4; ++qd) {
            const float4 v = *(const float4*)(xnp + 4 * qd);
            xn[4*qd+0] = v.x; xn[4*qd+1] = v.y; xn[4*qd+2] = v.z; xn[4*qd+3] = v.w;
        }

        // ================= layer 0 gates =================
        v8f acc0, acc1;
        #pragma unroll
        for (int r = 0; r < 8; ++r) { acc0[r] = b0a[r]; acc1[r] = b0b[r]; }

        #pragma unroll
        for (int kc = 0; kc < 8; ++kc) {
            v2f bx;
            bx.x = xc[kc * 2];
            bx.y = xc[kc * 2 + 1];
            acc0 = wmma4(Aih0[0][kc], bx, acc0);
            acc1 = wmma4(Aih0[1][kc], bx, acc1);
        }
        {   // recurrent from h1 (previous step) — lane-local B
            v2f bh;
            bh.x = h1l[0]; bh.y = h1l[1];
            acc0 = wmma4(Ahh0[0][0], bh, acc0);
            acc1 = wmma4(Ahh0[1][0], bh, acc1);
            bh.x = h1l[2]; bh.y = 0.0f;
            acc0 = wmma4(Ahh0[0][1], bh, acc0);
            acc1 = wmma4(Ahh0[1][1], bh, acc1);
        }
        {   // activations: this lane's 3 hidden units, all gates local
            const float iv[3] = { acc0[0], acc0[4], acc1[0] };
            const float fv[3] = { acc0[1], acc0[5], acc1[1] };
            const float gv[3] = { acc0[2], acc0[6], acc1[2] };
            const float ov[3] = { acc0[3], acc0[7], acc1[3] };
            #pragma unroll
            for (int kk = 0; kk < 3; ++kk) {
                const float ik = sigm(iv[kk]);
                const float fk = sigm(fv[kk]);
                const float gk = tanh_f(gv[kk]);
                const float ok = sigm(ov[kk]);
                c1l[kk] = fk * c1l[kk] + ik * gk;
                h1l[kk] = ok * tanh_f(c1l[kk]);
            }
        }

        // ================= layer 1 gates =================
        v8f d0, d1;
        #pragma unroll
        for (int r = 0; r < 8; ++r) { d0[r] = b1a[r]; d1[r] = b1b[r]; }
        {
            v2f bh;
            bh.x = h1l[0]; bh.y = h1l[1];        // input proj from h1 (current)
            d0 = wmma4(Aih1[0][0], bh, d0);
            d1 = wmma4(Aih1[1][0], bh, d1);
            bh.x = h1l[2]; bh.y = 0.0f;
            d0 = wmma4(Aih1[0][1], bh, d0);
            d1 = wmma4(Aih1[1][1], bh, d1);
            bh.x = h2l[0]; bh.y = h2l[1];        // recurrent from h2 (previous)
            d0 = wmma4(Ahh1[0][0], bh, d0);
            d1 = wmma4(Ahh1[1][0], bh, d1);
            bh.x = h2l[2]; bh.y = 0.0f;
            d0 = wmma4(Ahh1[0][1], bh, d0);
            d1 = wmma4(Ahh1[1][1], bh, d1);
        }
        {
            const float iv[3] = { d0[0], d0[4], d1[0] };
            const float fv[3] = { d0[1], d0[5], d1[1] };
            const float gv[3] = { d0[2], d0[6], d1[2] };
            const float ov[3] = { d0[3], d0[7], d1[3] };
            #pragma unroll
            for (int kk = 0; kk < 3; ++kk) {
                const float ik = sigm(iv[kk]);
                const float fk = sigm(fv[kk]);
                const float gk = tanh_f(gv[kk]);
                const float ok = sigm(ov[kk]);
                c2l[kk] = fk * c2l[kk] + ik * gk;
                h2l[kk] = ok * tanh_f(c2l[kk]);
            }
        }

        // rotate x double-buffer
        #pragma unroll
        for (int e = 0; e < 16; ++e) xc[e] = xn[e];
    }

    // Linear head: partial dot over this lane's 3 hidden units, one cross-half add.
    float part = h2l[0] * Wlin[3 * hi + 0]
               + h2l[1] * Wlin[3 * hi + 1]
               + h2l[2] * Wlin[3 * hi + 2];
    const float other = __shfl_xor(part, 16, 32);
    if (hi == 0) {
        out[bglob] = part + other + blin[0];
    }
}

extern "C" void kernel_launch(void* const* d_in, const int* in_sizes, int n_in,
                              void* d_out, int out_size, void* d_ws, size_t ws_size,
                              hipStream_t stream) {
    (void)in_sizes; (void)n_in; (void)out_size; (void)d_ws; (void)ws_size;
    const float* x    = (const float*)d_in[0];
    const float* Wih0 = (const float*)d_in[1];
    const float* Whh0 = (const float*)d_in[2];
    const float* bih0 = (const float*)d_in[3];
    const float* bhh0 = (const float*)d_in[4];
    const float* Wih1 = (const float*)d_in[5];
    const float* Whh1 = (const float*)d_in[6];
    const float* bih1 = (const float*)d_in[7];
    const float* bhh1 = (const float*)d_in[8];
    const float* Wlin = (const float*)d_in[9];
    const float* blin = (const float*)d_in[10];
    float* out = (float*)d_out;

    dim3 grid(TB / 16);   // 256 waves, 16 batch elements each
    dim3 block(32);       // one wave32 per block
    lstm_fused_kernel<<<grid, block, 0, stream>>>(x, Wih0, Whh0, bih0, bhh0,
                                                  Wih1, Whh1, bih1, bhh1,
                                                  Wlin, blin, out);
}